// NeuralSymplecticForm_62629213110311
// MI455X (gfx1250) — compile-verified
//
#include <hip/hip_runtime.h>

typedef float v2f __attribute__((ext_vector_type(2)));
typedef float v8f __attribute__((ext_vector_type(8)));

#define HD   128   // hidden width
#define BT   16    // samples per workgroup
#define NJ   64    // jacobian columns = 4*BT
#define LDA  132   // padded LDS stride (bank-conflict free)

static __device__ __forceinline__ v8f vzero8() {
  v8f z = {0.f,0.f,0.f,0.f,0.f,0.f,0.f,0.f};
  return z;
}

static __device__ __forceinline__ v8f wmma_k4(v2f a, v2f b, v8f c) {
  // D = A(16x4 f32) * B(4x16 f32) + C(16x16 f32)
  return __builtin_amdgcn_wmma_f32_16x16x4_f32(false, a, false, b, (short)0, c, false, false);
}

__global__ __launch_bounds__(256) void nsf_fused(
    const float* __restrict__ u,
    const float* __restrict__ fw1, const float* __restrict__ fb1,
    const float* __restrict__ fw2, const float* __restrict__ fb2,
    const float* __restrict__ fw3, const float* __restrict__ fb3,
    const float* __restrict__ fw4, const float* __restrict__ fb4,
    const float* __restrict__ gw1, const float* __restrict__ gb1,
    const float* __restrict__ gw2, const float* __restrict__ gb2,
    const float* __restrict__ gw3, const float* __restrict__ gb3,
    const float* __restrict__ gw4, const float* __restrict__ gb4,
    float* __restrict__ out)
{
  __shared__ float sact[BT * LDA];   // activations: [sample][feature] (padded)
  __shared__ float sjac[NJ * LDA];   // jacobian chain: [col=(b*4+k)][feature] (padded)
  __shared__ float smask[HD * BT];   // relu mask of one layer: [feature][sample]
  __shared__ float sJ[BT * 16];      // per-sample 4x4 jacobian
  __shared__ float sgrad[BT * 4];    // per-sample grad_H

  const int tid  = threadIdx.x;
  const int lane = tid & 31;
  const int wid  = tid >> 5;          // 0..7, each wave owns 16 feature rows
  const int hi   = lane >> 4;         // lane half (WMMA K split)
  const int ln   = lane & 15;
  const int m0   = wid * 16;
  const long long s0 = (long long)blockIdx.x * BT;

  (void)fb4; // unused: output-layer bias does not enter the Jacobian

  // ---------------- P0: f layer1  h1 = fw1 @ u^T  (K = 4, single WMMA) -----
  {
    v8f c = vzero8();
    v2f a, b;
    const int kb = 2 * hi;
    a.x = fw1[(m0 + ln) * 4 + kb];     a.y = fw1[(m0 + ln) * 4 + kb + 1];
    b.x = u[(s0 + ln) * 4 + kb];       b.y = u[(s0 + ln) * 4 + kb + 1];
    c = wmma_k4(a, b, c);
#pragma unroll
    for (int r = 0; r < 8; ++r) {
      const int row = m0 + r + 8 * hi;
      const float h = c[r] + fb1[row];
      const float m = (h > 0.f) ? 1.f : 0.f;
      smask[row * BT + ln] = m;          // m1
      sact[ln * LDA + row] = h * m;      // a1f
    }
  }
  __syncthreads();

  // ---------------- P1: build J1 into sjac (uses m1);  f layer2 GEMM -------
#pragma unroll
  for (int i = 0; i < (NJ * HD) / 256; ++i) {
    const int e   = tid + 256 * i;
    const int col = e >> 7;             // b*4 + k
    const int j   = e & (HD - 1);
    sjac[col * LDA + j] = smask[j * BT + (col >> 2)] * fw1[j * 4 + (col & 3)];
  }
  v8f hreg = vzero8();
#pragma unroll
  for (int kk = 0; kk < HD / 4; ++kk) {
    const int kb = kk * 4 + 2 * hi;
    v2f a, b;
    a.x = fw2[(m0 + ln) * HD + kb];  a.y = fw2[(m0 + ln) * HD + kb + 1];
    b.x = sact[ln * LDA + kb];       b.y = sact[ln * LDA + kb + 1];
    hreg = wmma_k4(a, b, hreg);
  }
  __syncthreads();

  // ---------------- P2: commit a2f and m2 ----------------------------------
#pragma unroll
  for (int r = 0; r < 8; ++r) {
    const int row = m0 + r + 8 * hi;
    const float h = hreg[r] + fb2[row];
    const float m = (h > 0.f) ? 1.f : 0.f;
    smask[row * BT + ln] = m;            // m2
    sact[ln * LDA + row] = h * m;        // a2f
  }
  __syncthreads();

  // ---------------- P3: Y2 = fw2 @ J1 (regs);  f layer3 GEMM (regs) --------
  v8f y[4];
#pragma unroll
  for (int nt = 0; nt < 4; ++nt) y[nt] = vzero8();
#pragma unroll 4
  for (int kk = 0; kk < HD / 4; ++kk) {
    const int kb = kk * 4 + 2 * hi;
    v2f a;
    a.x = fw2[(m0 + ln) * HD + kb];  a.y = fw2[(m0 + ln) * HD + kb + 1];
#pragma unroll
    for (int nt = 0; nt < 4; ++nt) {
      v2f b;
      const int col = nt * 16 + ln;
      b.x = sjac[col * LDA + kb];    b.y = sjac[col * LDA + kb + 1];
      y[nt] = wmma_k4(a, b, y[nt]);
    }
  }
  v8f h3 = vzero8();
#pragma unroll
  for (int kk = 0; kk < HD / 4; ++kk) {
    const int kb = kk * 4 + 2 * hi;
    v2f a, b;
    a.x = fw3[(m0 + ln) * HD + kb];  a.y = fw3[(m0 + ln) * HD + kb + 1];
    b.x = sact[ln * LDA + kb];       b.y = sact[ln * LDA + kb + 1];
    h3 = wmma_k4(a, b, h3);
  }
  __syncthreads();

  // ---------------- P4: masked commit  sjac = m2 .* Y2 ----------------------
#pragma unroll
  for (int nt = 0; nt < 4; ++nt) {
#pragma unroll
    for (int r = 0; r < 8; ++r) {
      const int j   = m0 + r + 8 * hi;
      const int col = nt * 16 + ln;
      sjac[col * LDA + j] = y[nt][r] * smask[j * BT + (col >> 2)];
    }
  }
  __syncthreads();

  // ---------------- P5: commit m3 -------------------------------------------
#pragma unroll
  for (int r = 0; r < 8; ++r) {
    const int row = m0 + r + 8 * hi;
    const float h = h3[r] + fb3[row];
    smask[row * BT + ln] = (h > 0.f) ? 1.f : 0.f;   // m3
  }
  __syncthreads();

  // ---------------- P6: Y3 = fw3 @ Y2 (regs) --------------------------------
#pragma unroll
  for (int nt = 0; nt < 4; ++nt) y[nt] = vzero8();
#pragma unroll 4
  for (int kk = 0; kk < HD / 4; ++kk) {
    const int kb = kk * 4 + 2 * hi;
    v2f a;
    a.x = fw3[(m0 + ln) * HD + kb];  a.y = fw3[(m0 + ln) * HD + kb + 1];
#pragma unroll
    for (int nt = 0; nt < 4; ++nt) {
      v2f b;
      const int col = nt * 16 + ln;
      b.x = sjac[col * LDA + kb];    b.y = sjac[col * LDA + kb + 1];
      y[nt] = wmma_k4(a, b, y[nt]);
    }
  }
  __syncthreads();

  // ---------------- P7: masked commit  sjac = m3 .* Y3 ----------------------
#pragma unroll
  for (int nt = 0; nt < 4; ++nt) {
#pragma unroll
    for (int r = 0; r < 8; ++r) {
      const int j   = m0 + r + 8 * hi;
      const int col = nt * 16 + ln;
      sjac[col * LDA + j] = y[nt][r] * smask[j * BT + (col >> 2)];
    }
  }
  __syncthreads();

  // ---------------- P8: J = fw4 @ Y3 (thin, VALU);  g layer1 WMMA -----------
  {
    v8f c = vzero8();
    v2f a, b;
    const int kb = 2 * hi;
    a.x = gw1[(m0 + ln) * 4 + kb];   a.y = gw1[(m0 + ln) * 4 + kb + 1];
    b.x = u[(s0 + ln) * 4 + kb];     b.y = u[(s0 + ln) * 4 + kb + 1];
    c = wmma_k4(a, b, c);
#pragma unroll
    for (int r = 0; r < 8; ++r) {
      const int row = m0 + r + 8 * hi;
      const float h = c[r] + gb1[row];
      sact[ln * LDA + row] = (h > 0.f) ? h : 0.f;   // a1g
    }
  }
  {
    // 256 threads == 16 samples * 16 J-entries: one dot product each
    const int b = tid >> 4;
    const int o = (tid >> 2) & 3;
    const int k = tid & 3;
    const float* __restrict__ colp = &sjac[(b * 4 + k) * LDA];
    const float* __restrict__ w4r  = &fw4[o * HD];
    float acc = 0.f;
#pragma unroll 8
    for (int j = 0; j < HD; ++j) acc += w4r[j] * colp[j];
    sJ[tid] = acc;                      // sJ[b*16 + o*4 + k] = J[o][k]
  }
  __syncthreads();

  // ---------------- P9: g layer2 --------------------------------------------
  hreg = vzero8();
#pragma unroll
  for (int kk = 0; kk < HD / 4; ++kk) {
    const int kb = kk * 4 + 2 * hi;
    v2f a, b;
    a.x = gw2[(m0 + ln) * HD + kb];  a.y = gw2[(m0 + ln) * HD + kb + 1];
    b.x = sact[ln * LDA + kb];       b.y = sact[ln * LDA + kb + 1];
    hreg = wmma_k4(a, b, hreg);
  }
  __syncthreads();
#pragma unroll
  for (int r = 0; r < 8; ++r) {
    const int row = m0 + r + 8 * hi;
    const float h = hreg[r] + gb2[row];
    sact[ln * LDA + row] = (h > 0.f) ? h : 0.f;     // a2g
  }
  __syncthreads();

  // ---------------- P10: g layer3 -------------------------------------------
  hreg = vzero8();
#pragma unroll
  for (int kk = 0; kk < HD / 4; ++kk) {
    const int kb = kk * 4 + 2 * hi;
    v2f a, b;
    a.x = gw3[(m0 + ln) * HD + kb];  a.y = gw3[(m0 + ln) * HD + kb + 1];
    b.x = sact[ln * LDA + kb];       b.y = sact[ln * LDA + kb + 1];
    hreg = wmma_k4(a, b, hreg);
  }
  __syncthreads();
#pragma unroll
  for (int r = 0; r < 8; ++r) {
    const int row = m0 + r + 8 * hi;
    const float h = hreg[r] + gb3[row];
    sact[ln * LDA + row] = (h > 0.f) ? h : 0.f;     // a3g
  }
  __syncthreads();

  // ---------------- P11: grad_H = gw4 @ a3g + gb4 (thin, VALU) --------------
  if (tid < BT * 4) {
    const int b = tid >> 2;
    const int o = tid & 3;
    const float* __restrict__ w4r = &gw4[o * HD];
    const float* __restrict__ ap  = &sact[b * LDA];
    float acc = gb4[o];
#pragma unroll 8
    for (int j = 0; j < HD; ++j) acc += w4r[j] * ap[j];
    sgrad[tid] = acc;
  }
  __syncthreads();

  // ---------------- P12: exact 4x4 antisymmetric solve (Pfaffian inverse) ---
  // A = J - J^T (antisymmetric);  A^{-1} = (1/Pf) * [[0,-f,e,-d],[f,0,-c,b],
  //                                                  [-e,c,0,-a],[d,-b,a,0]]
  if (tid < BT) {
    const float* __restrict__ Jb = &sJ[tid * 16];
    const float a01 = Jb[1]  - Jb[4];
    const float a02 = Jb[2]  - Jb[8];
    const float a03 = Jb[3]  - Jb[12];
    const float a12 = Jb[6]  - Jb[9];
    const float a13 = Jb[7]  - Jb[13];
    const float a23 = Jb[11] - Jb[14];
    const float g0 = sgrad[tid * 4 + 0];
    const float g1 = sgrad[tid * 4 + 1];
    const float g2 = sgrad[tid * 4 + 2];
    const float g3 = sgrad[tid * 4 + 3];
    const float pf  = a01 * a23 - a02 * a13 + a03 * a12;
    const float inv = 1.f / pf;
    float* __restrict__ op = &out[(s0 + tid) * 4];
    op[0] = (-a23 * g1 + a13 * g2 - a12 * g3) * inv;
    op[1] = ( a23 * g0 - a03 * g2 + a02 * g3) * inv;
    op[2] = (-a13 * g0 + a03 * g1 - a01 * g3) * inv;
    op[3] = ( a12 * g0 - a02 * g1 + a01 * g2) * inv;
  }
}

extern "C" void kernel_launch(void* const* d_in, const int* in_sizes, int n_in,
                              void* d_out, int out_size, void* d_ws, size_t ws_size,
                              hipStream_t stream) {
  (void)n_in; (void)d_ws; (void)ws_size; (void)out_size;
  const float* u   = (const float*)d_in[0];
  const float* fw1 = (const float*)d_in[1];
  const float* fb1 = (const float*)d_in[2];
  const float* fw2 = (const float*)d_in[3];
  const float* fb2 = (const float*)d_in[4];
  const float* fw3 = (const float*)d_in[5];
  const float* fb3 = (const float*)d_in[6];
  const float* fw4 = (const float*)d_in[7];
  const float* fb4 = (const float*)d_in[8];
  const float* gw1 = (const float*)d_in[9];
  const float* gb1 = (const float*)d_in[10];
  const float* gw2 = (const float*)d_in[11];
  const float* gb2 = (const float*)d_in[12];
  const float* gw3 = (const float*)d_in[13];
  const float* gb3 = (const float*)d_in[14];
  const float* gw4 = (const float*)d_in[15];
  const float* gb4 = (const float*)d_in[16];
  float* out = (float*)d_out;

  const int B = in_sizes[0] / 4;       // 131072
  const int grid = B / BT;             // 8192 workgroups, 16 samples each
  nsf_fused<<<grid, 256, 0, stream>>>(u,
      fw1, fb1, fw2, fb2, fw3, fb3, fw4, fb4,
      gw1, gb1, gw2, gb2, gw3, gb3, gw4, gb4,
      out);
}